// DVSEventProcessor_44092134260933
// MI455X (gfx1250) — compile-verified
//
#include <hip/hip_runtime.h>

// DVS event -> spike tensor scatter. Shape (2, 346, 260, time_steps), f32.
// Memory-bound: ~230 MB moved -> ~10us at 23.3 TB/s on MI455X.
//
// MI455X-specific strategy:
//  * The 72 MB output fits in the 192 MB global L2 -> zero it with default
//    (RT) stores so it is L2-resident, then let the 8M random 4B scatter
//    stores hit L2 instead of doing HBM read-modify-write.
//  * All streaming reads (160 MB of events + timestamps) use non-temporal
//    (TH=NT) loads so they do NOT evict the L2-resident output tensor.
//  * wave32 shuffle reduction for the timestamp min/max (gfx1250 is wave32).
//  * global_prefetch_b8 one grid-stride ahead on the streaming loops.
//  * No WMMA on purpose: this is a scatter histogram, there is no matrix
//    tile structure to feed the WMMA pipes.

#define SENSOR_W 346
#define SENSOR_H 260
#define WAVE 32  // gfx1250 is wave32-only

// Native clang vector types: required by __builtin_nontemporal_load (HIP's
// float4/int4 are structs and are rejected).
typedef __attribute__((ext_vector_type(4))) float vf4;
typedef __attribute__((ext_vector_type(4))) int   vi4;

// ---- order-preserving float<->uint transform (handles negatives too) ----
__device__ __forceinline__ unsigned f2ord(float f) {
    unsigned b = __float_as_uint(f);
    return (b & 0x80000000u) ? ~b : (b | 0x80000000u);
}
__device__ __forceinline__ float ord2f(unsigned u) {
    return (u & 0x80000000u) ? __uint_as_float(u & 0x7FFFFFFFu)
                             : __uint_as_float(~u);
}

// ---------------- Kernel 1: zero output, seed ws[0]=min ws[1]=max ----------
// Default (RT) stores: we WANT the zeroed tensor resident/dirty in L2 so the
// scatter pass hits L2.
__global__ void dvs_init_kernel(float* __restrict__ out, int out_n,
                                unsigned* __restrict__ ws) {
    int tid = blockIdx.x * blockDim.x + threadIdx.x;
    if (tid == 0) {
        ws[0] = 0xFFFFFFFFu;  // min identity in ordered-uint space
        ws[1] = 0u;           // max identity
    }
    int stride = gridDim.x * blockDim.x;
    int n4 = out_n >> 2;
    vf4* o4 = reinterpret_cast<vf4*>(out);
    vf4 z = (vf4)(0.f);
    for (int k = tid; k < n4; k += stride) o4[k] = z;
    for (int k = (n4 << 2) + tid; k < out_n; k += stride) out[k] = 0.f;  // tail
}

// ---------------- Kernel 2: global min/max of t (wave32 reduction) ---------
__global__ void dvs_minmax_kernel(const float* __restrict__ t, int n,
                                  unsigned* __restrict__ ws) {
    unsigned lmin = 0xFFFFFFFFu, lmax = 0u;
    int tid = blockIdx.x * blockDim.x + threadIdx.x;
    int stride = gridDim.x * blockDim.x;
    int n4 = n >> 2;
    const vf4* t4 = reinterpret_cast<const vf4*>(t);
    for (int k = tid; k < n4; k += stride) {
        if (k + stride < n4)
            __builtin_prefetch(&t4[k + stride], 0, 0);  // global_prefetch_b8
        vf4 v = __builtin_nontemporal_load(&t4[k]);     // TH=NT load
        unsigned a = f2ord(v.x), b = f2ord(v.y), c = f2ord(v.z), d = f2ord(v.w);
        lmin = min(min(lmin, a), min(min(b, c), d));
        lmax = max(max(lmax, a), max(max(b, c), d));
    }
    for (int k = (n4 << 2) + tid; k < n; k += stride) {  // tail
        unsigned a = f2ord(__builtin_nontemporal_load(&t[k]));
        lmin = min(lmin, a);
        lmax = max(lmax, a);
    }
    // wave32 butterfly reduction
    #pragma unroll
    for (int off = WAVE / 2; off > 0; off >>= 1) {
        lmin = min(lmin, (unsigned)__shfl_xor((int)lmin, off, WAVE));
        lmax = max(lmax, (unsigned)__shfl_xor((int)lmax, off, WAVE));
    }
    __shared__ unsigned smin[8], smax[8];  // 256 threads = 8 waves
    int wave = threadIdx.x / WAVE, lane = threadIdx.x % WAVE;
    if (lane == 0) { smin[wave] = lmin; smax[wave] = lmax; }
    __syncthreads();
    if (threadIdx.x == 0) {
        int nw = blockDim.x / WAVE;
        unsigned m0 = smin[0], m1 = smax[0];
        for (int w = 1; w < nw; ++w) {
            m0 = min(m0, smin[w]);
            m1 = max(m1, smax[w]);
        }
        atomicMin(&ws[0], m0);
        atomicMax(&ws[1], m1);
    }
}

// ---------------- Kernel 3: scatter 1.0f per valid event ------------------
// All valid events write the identical value 1.0f (reference: .max(valid)
// onto a zeroed tensor), so plain stores suffice -- no atomics; races write
// the same value. Invalid events write nothing (max with 0 is a no-op).
// Event streams are loaded non-temporally (4 events/thread, b128 loads) so
// the 128 MB stream does not evict the L2-resident output.
__global__ void dvs_scatter_kernel(const int* __restrict__ x,
                                   const int* __restrict__ y,
                                   const float* __restrict__ t,
                                   const int* __restrict__ p,
                                   const int* __restrict__ time_steps_p,
                                   const unsigned* __restrict__ ws,
                                   float* __restrict__ out, int n) {
    const int ts = *time_steps_p;  // uniform scalar load
    const float tmin = ord2f(ws[0]);
    const float tmax = ord2f(ws[1]);
    const bool has_range = tmax > tmin;
    const float denom = has_range ? (tmax - tmin) : 1.0f;
    const float scale = (float)(ts - 1);  // ref: ((t-tmin)/denom)*(ts-1)

    int tid = blockIdx.x * blockDim.x + threadIdx.x;
    int stride = gridDim.x * blockDim.x;
    int n4 = n >> 2;
    const vi4* x4 = reinterpret_cast<const vi4*>(x);
    const vi4* y4 = reinterpret_cast<const vi4*>(y);
    const vf4* t4 = reinterpret_cast<const vf4*>(t);
    const vi4* p4 = reinterpret_cast<const vi4*>(p);

    for (int k = tid; k < n4; k += stride) {
        if (k + stride < n4) {  // gfx1250 global_prefetch_b8, one stride ahead
            __builtin_prefetch(&x4[k + stride], 0, 0);
            __builtin_prefetch(&y4[k + stride], 0, 0);
            __builtin_prefetch(&t4[k + stride], 0, 0);
            __builtin_prefetch(&p4[k + stride], 0, 0);
        }
        vi4 xv = __builtin_nontemporal_load(&x4[k]);  // TH=NT b128 loads
        vi4 yv = __builtin_nontemporal_load(&y4[k]);
        vf4 tv = __builtin_nontemporal_load(&t4[k]);
        vi4 pv = __builtin_nontemporal_load(&p4[k]);

        #pragma unroll
        for (int e = 0; e < 4; ++e) {
            int   xe = xv[e];
            int   ye = yv[e];
            int   pe = pv[e];
            float te = tv[e];
            int tn = 0;
            if (has_range) tn = (int)(((te - tmin) / denom) * scale);
            bool valid = (xe >= 0) & (xe < SENSOR_W) & (ye >= 0) &
                         (ye < SENSOR_H) & (tn >= 0) & (tn < ts);
            if (valid) {
                int ch = pe < 0 ? 0 : (pe > 1 ? 1 : pe);
                // layout (2, W, H, T) row-major; fits in 32-bit
                int idx = ((ch * SENSOR_W + xe) * SENSOR_H + ye) * ts + tn;
                out[idx] = 1.0f;  // L2-resident scatter (72 MB < 192 MB L2)
            }
        }
    }
    // scalar tail (n % 4)
    for (int k = (n4 << 2) + tid; k < n; k += stride) {
        int xv = x[k], yv = y[k], pv = p[k];
        float tv = __builtin_nontemporal_load(&t[k]);
        int tn = 0;
        if (has_range) tn = (int)(((tv - tmin) / denom) * scale);
        bool valid = (xv >= 0) & (xv < SENSOR_W) & (yv >= 0) & (yv < SENSOR_H) &
                     (tn >= 0) & (tn < ts);
        if (valid) {
            int ch = pv < 0 ? 0 : (pv > 1 ? 1 : pv);
            int idx = ((ch * SENSOR_W + xv) * SENSOR_H + yv) * ts + tn;
            out[idx] = 1.0f;
        }
    }
}

extern "C" void kernel_launch(void* const* d_in, const int* in_sizes, int n_in,
                              void* d_out, int out_size, void* d_ws,
                              size_t ws_size, hipStream_t stream) {
    const int*   x  = (const int*)d_in[0];
    const int*   y  = (const int*)d_in[1];
    const float* t  = (const float*)d_in[2];
    const int*   p  = (const int*)d_in[3];
    const int*   ts = (const int*)d_in[4];  // scalar time_steps
    float*    out = (float*)d_out;
    unsigned* ws  = (unsigned*)d_ws;  // ws[0]=ordered-min, ws[1]=ordered-max

    const int n = in_sizes[0];
    const int threads = 256;  // 8 wave32s per block

    // 1) zero 72 MB output (RT stores -> L2-resident) + seed min/max cells
    dvs_init_kernel<<<4096, threads, 0, stream>>>(out, out_size, ws);
    // 2) min/max over 8M timestamps (NT b128 loads, wave32 shuffle tree)
    dvs_minmax_kernel<<<2048, threads, 0, stream>>>(t, n, ws);
    // 3) stream events (NT b128 loads, 4 events/thread, ~2 iters/thread)
    dvs_scatter_kernel<<<4096, threads, 0, stream>>>(x, y, t, p, ts, ws, out, n);
}